// SegmentKeypointModel_9070970929711
// MI455X (gfx1250) — compile-verified
//
#include <hip/hip_runtime.h>
#include <hip/hip_bf16.h>

// Problem constants (match reference setup_inputs)
#define HH 512
#define WW 512
#define NUM_KP 17
#define BATCH 16
#define PLANES (BATCH * NUM_KP)            // 272
#define TILE_ROWS 8
#define TILES_PER_PLANE (HH / TILE_ROWS)   // 64
#define TOPK 30
#define THRESH 0.1f
#define LROW 520                           // padded LDS row stride in floats (bank skew, 16B aligned)
#define NTHREADS 256
#define CAND (TILES_PER_PLANE * TOPK)      // 1920 candidates per plane in phase 2

#if defined(__gfx1250__) && __has_builtin(__builtin_amdgcn_global_load_async_to_lds_b128)
#define USE_ASYNC_LDS 1
#else
#define USE_ASYNC_LDS 0
#endif

typedef int v4i __attribute__((ext_vector_type(4)));
typedef __attribute__((address_space(1))) v4i* gas_v4i_ptr;   // global AS (prints as __device__)
typedef __attribute__((address_space(3))) v4i* las_v4i_ptr;   // LDS AS (prints as __shared__)

__device__ __forceinline__ unsigned long long u64max(unsigned long long a, unsigned long long b) {
    return a > b ? a : b;
}

// 64-bit xor-shuffle built from two 32-bit shuffles (wave32)
__device__ __forceinline__ unsigned long long shfl_xor_u64(unsigned long long v, int m) {
    int lo = __shfl_xor((int)(unsigned int)v, m, 32);
    int hi = __shfl_xor((int)(unsigned int)(v >> 32), m, 32);
    return ((unsigned long long)(unsigned int)hi << 32) | (unsigned long long)(unsigned int)lo;
}

// Monotonic key: higher score wins; equal scores -> smaller pixel index wins (matches lax.top_k)
__device__ __forceinline__ unsigned long long packkey(float s, unsigned int pix) {
    return ((unsigned long long)__float_as_uint(s) << 32) |
           (unsigned long long)(0xFFFFFFFFu - pix);
}

// ---------------------------------------------------------------------------
// Phase 1: per (plane, 8-row tile) -> exact top-30 packed keys of NMS peaks
// ---------------------------------------------------------------------------
__global__ __launch_bounds__(NTHREADS)
void pose_nms_tile_topk(const float* __restrict__ heat,
                        unsigned long long* __restrict__ blockkeys) {
    __shared__ float tile[(TILE_ROWS + 2) * LROW];          // 10 rows (incl. halo)
    __shared__ float sc[TILE_ROWS * WW];                    // peak score per pixel (0 = invalid)
    __shared__ unsigned long long wred[NTHREADS / 32];

    const int tid  = threadIdx.x;
    const int lane = tid & 31;
    const int ry   = tid >> 5;          // wave id == tile row 0..7
    const int bi   = blockIdx.x;
    const int p    = bi / TILES_PER_PLANE;
    const int t    = bi % TILES_PER_PLANE;
    const int y0   = t * TILE_ROWS;
    const float* plane = heat + (size_t)p * (HH * WW);
    const float NEG_INF = -__builtin_inff();

    // Warm L2 for the next tile of this plane (gfx1250 global_prefetch_b8)
    {
        int ny0 = (y0 + TILE_ROWS < HH) ? (y0 + TILE_ROWS) : 0;
        __builtin_prefetch(plane + (size_t)ny0 * WW + tid * 16, 0, 1);
    }

    // Out-of-plane halo rows -> -inf
    if (y0 == 0)
        for (int i = tid; i < WW; i += NTHREADS) tile[0 * LROW + i] = NEG_INF;
    if (y0 + TILE_ROWS == HH)
        for (int i = tid; i < WW; i += NTHREADS) tile[(TILE_ROWS + 1) * LROW + i] = NEG_INF;

    // Stage 10 rows x 512 floats into LDS as 16B vectors (1280 vec4 = 5 per thread)
    const int NV = (TILE_ROWS + 2) * (WW / 4);
    for (int k = 0; k < NV / NTHREADS; ++k) {
        int i  = tid + k * NTHREADS;
        int r  = i >> 7;               // / (WW/4)
        int cv = i & (WW / 4 - 1);
        int gy = y0 - 1 + r;
        if (gy >= 0 && gy < HH) {
            const float* g = plane + (size_t)gy * WW + cv * 4;
            float* l = &tile[r * LROW + cv * 4];
#if USE_ASYNC_LDS
            __builtin_amdgcn_global_load_async_to_lds_b128(
                (gas_v4i_ptr)g, (las_v4i_ptr)l, 0, 0);
#else
            *(float4*)l = *(const float4*)g;
#endif
        }
    }
#if USE_ASYNC_LDS
#if __has_builtin(__builtin_amdgcn_s_wait_asynccnt)
    __builtin_amdgcn_s_wait_asynccnt(0);
#else
    asm volatile("s_wait_asynccnt 0" ::: "memory");
#endif
#endif
    __syncthreads();

    // ---- 3x3 NMS, lane-interleaved vector layout (conflict-free LDS) ----
    // Wave ry handles tile row ry; lane owns cols c0 = 4*lane + 128*i, i=0..3.
    float m[4][4];    // vertical 3-max per chunk/col
    float ce[4][4];   // center-row values
#pragma unroll
    for (int i = 0; i < 4; ++i) {
        int c0 = 4 * lane + 128 * i;
        float4 a = *(const float4*)&tile[(ry + 0) * LROW + c0];
        float4 b = *(const float4*)&tile[(ry + 1) * LROW + c0];
        float4 d = *(const float4*)&tile[(ry + 2) * LROW + c0];
        ce[i][0] = b.x; ce[i][1] = b.y; ce[i][2] = b.z; ce[i][3] = b.w;
        m[i][0] = fmaxf(fmaxf(a.x, b.x), d.x);
        m[i][1] = fmaxf(fmaxf(a.y, b.y), d.y);
        m[i][2] = fmaxf(fmaxf(a.z, b.z), d.z);
        m[i][3] = fmaxf(fmaxf(a.w, b.w), d.w);
    }

    unsigned long long tmax = 0ull;
#pragma unroll
    for (int i = 0; i < 4; ++i) {
        // neighbor columns across lanes / chunks via wave32 shuffles
        float lm = __shfl_up(m[i][3], 1, 32);                            // lane-1's col c0-1
        float lmp = (i > 0) ? __shfl(m[i - 1][3], 31, 32) : NEG_INF;     // prev chunk, lane31
        if (lane == 0) lm = lmp;
        float rm = __shfl_down(m[i][0], 1, 32);                          // lane+1's col c0+4
        float rmn = (i < 3) ? __shfl(m[i + 1][0], 0, 32) : NEG_INF;      // next chunk, lane0
        if (lane == 31) rm = rmn;

        float hh[4];
        hh[0] = fmaxf(fmaxf(lm,      m[i][0]), m[i][1]);
        hh[1] = fmaxf(fmaxf(m[i][0], m[i][1]), m[i][2]);
        hh[2] = fmaxf(fmaxf(m[i][1], m[i][2]), m[i][3]);
        hh[3] = fmaxf(fmaxf(m[i][2], m[i][3]), rm);

        int c0 = 4 * lane + 128 * i;
        unsigned int pixb = (unsigned int)((y0 + ry) * WW + c0);
        float sv[4];
#pragma unroll
        for (int j = 0; j < 4; ++j) {
            float cv = ce[i][j];
            bool keep = (cv == hh[j]) && (cv > THRESH);   // equals 3x3 max AND above threshold
            sv[j] = keep ? cv : 0.0f;
            if (keep) tmax = u64max(tmax, packkey(cv, pixb + j));
        }
        float4 st; st.x = sv[0]; st.y = sv[1]; st.z = sv[2]; st.w = sv[3];
        *(float4*)&sc[ry * WW + c0] = st;
    }

    // ---- exact tile top-30 by repeated max reduction with removal ----
    unsigned long long* outk = blockkeys + (size_t)bi * TOPK;
    for (int r = 0; r < TOPK; ++r) {
        unsigned long long w = tmax;
#pragma unroll
        for (int mm = 16; mm >= 1; mm >>= 1) w = u64max(w, shfl_xor_u64(w, mm));
        if (lane == 0) wred[ry] = w;
        __syncthreads();
        unsigned long long g = wred[0];
#pragma unroll
        for (int k = 1; k < NTHREADS / 32; ++k) g = u64max(g, wred[k]);
        if (tid == 0) outk[r] = g;
        if (g != 0ull && tmax == g) {        // unique owner (keys embed pixel index)
            unsigned long long nm = 0ull;
#pragma unroll
            for (int i = 0; i < 4; ++i) {
                int c0 = 4 * lane + 128 * i;
                float* sp = &sc[ry * WW + c0];
                float4 q = *(float4*)sp;
                float qq[4] = {q.x, q.y, q.z, q.w};
                unsigned int pixb = (unsigned int)((y0 + ry) * WW + c0);
                bool ch = false;
#pragma unroll
                for (int j = 0; j < 4; ++j) {
                    unsigned long long key =
                        (qq[j] > THRESH) ? packkey(qq[j], pixb + j) : 0ull;
                    if (key == g) { qq[j] = 0.0f; ch = true; key = 0ull; }
                    nm = u64max(nm, key);
                }
                if (ch) {
                    float4 st; st.x = qq[0]; st.y = qq[1]; st.z = qq[2]; st.w = qq[3];
                    *(float4*)sp = st;
                }
            }
            tmax = nm;
        }
        __syncthreads();
    }
}

// ---------------------------------------------------------------------------
// Phase 2: per plane, merge 64 tiles x 30 keys -> final sorted top-30, decode
// ---------------------------------------------------------------------------
__global__ __launch_bounds__(NTHREADS)
void pose_merge_topk(const unsigned long long* __restrict__ blockkeys,
                     float* __restrict__ out) {
    __shared__ unsigned long long cand[CAND];
    __shared__ unsigned long long wred[NTHREADS / 32];
    __shared__ unsigned long long fin[TOPK];

    const int tid = threadIdx.x;
    const int p   = blockIdx.x;
    const unsigned long long* src = blockkeys + (size_t)p * CAND;
    for (int i = tid; i < CAND; i += NTHREADS) cand[i] = src[i];
    __syncthreads();

    unsigned long long tmax = 0ull;
#pragma unroll
    for (int k = 0; k < (CAND + NTHREADS - 1) / NTHREADS; ++k) {
        int i = tid + k * NTHREADS;
        if (i < CAND) tmax = u64max(tmax, cand[i]);
    }
    const int lane = tid & 31, wid = tid >> 5;
    for (int r = 0; r < TOPK; ++r) {
        unsigned long long w = tmax;
#pragma unroll
        for (int mm = 16; mm >= 1; mm >>= 1) w = u64max(w, shfl_xor_u64(w, mm));
        if (lane == 0) wred[wid] = w;
        __syncthreads();
        unsigned long long g = wred[0];
#pragma unroll
        for (int k = 1; k < NTHREADS / 32; ++k) g = u64max(g, wred[k]);
        if (tid == 0) fin[r] = g;
        if (g != 0ull && tmax == g) {
            unsigned long long nm = 0ull;
#pragma unroll
            for (int k = 0; k < (CAND + NTHREADS - 1) / NTHREADS; ++k) {
                int i = tid + k * NTHREADS;
                if (i < CAND) {
                    unsigned long long key = cand[i];
                    if (key == g) { cand[i] = 0ull; key = 0ull; }
                    nm = u64max(nm, key);
                }
            }
            tmax = nm;
        }
        __syncthreads();
    }

    // Decode and write outputs: kps [P,30,2], scores [P,30], valid [P,30] (concatenated)
    if (tid < TOPK) {
        unsigned long long k = fin[tid];
        float* out_kps = out;                           // PLANES*TOPK*2 floats
        float* out_sc  = out + PLANES * TOPK * 2;       // PLANES*TOPK floats
        float* out_va  = out + PLANES * TOPK * 3;       // PLANES*TOPK floats
        float score, kx, ky, va;
        if (k != 0ull) {
            score = __uint_as_float((unsigned int)(k >> 32));
            unsigned int pix = 0xFFFFFFFFu - (unsigned int)(k & 0xFFFFFFFFull);
            kx = (float)(4u * (pix % WW));
            ky = (float)(4u * (pix / WW));
            va = 1.0f;
        } else {
            score = -__builtin_inff();
            kx = 0.0f; ky = 0.0f; va = 0.0f;
        }
        size_t s = (size_t)p * TOPK + tid;
        out_kps[2 * s + 0] = kx;
        out_kps[2 * s + 1] = ky;
        out_sc[s] = score;
        out_va[s] = va;
    }
}

extern "C" void kernel_launch(void* const* d_in, const int* in_sizes, int n_in,
                              void* d_out, int out_size, void* d_ws, size_t ws_size,
                              hipStream_t stream) {
    (void)in_sizes; (void)n_in; (void)out_size; (void)ws_size;
    const float* heat = (const float*)d_in[0];
    unsigned long long* blockkeys = (unsigned long long*)d_ws;  // 272*64*30*8 = ~4.2 MB

    pose_nms_tile_topk<<<dim3(PLANES * TILES_PER_PLANE), dim3(NTHREADS), 0, stream>>>(
        heat, blockkeys);
    pose_merge_topk<<<dim3(PLANES), dim3(NTHREADS), 0, stream>>>(
        blockkeys, (float*)d_out);
}